// PointFPModule_1666447311445
// MI455X (gfx1250) — compile-verified
//
#include <hip/hip_runtime.h>

// ---------------------------------------------------------------------------
// PointNet++ feature propagation, fused for MI455X (gfx1250, wave32, WMMA).
//   stage 1: three_nn (idx + weights)            -> ws
//   stage 2: transpose source_feats (B,C2,m)->(B,m,C2) for coalesced gathers
//   stage 3: BN scale/bias precompute
//   stage 4: fused [interp+concat -> GEMM0(384->256)+BN+ReLU (LDS resident)
//                   -> GEMM1(256->256)+BN+ReLU -> staged coalesced store],
//            WMMA f32 16x16x4, K-outer loop: 1 LDS B-fragment feeds 4 WMMAs.
//            target_feats slab fill uses GLOBAL_LOAD_ASYNC_TO_LDS (ASYNCcnt).
// ---------------------------------------------------------------------------

#define B_    8
#define NPTS  16384
#define MPTS  1024
#define C1    128
#define C2    256
#define CIN   384
#define CO    256
#define NCOLS 32
#define XSTR  388   // CIN + 4 pad (bank-conflict-free stride, 8B aligned)
#define YSTR  260   // CO  + 4 pad

typedef float v2f __attribute__((ext_vector_type(2)));
typedef float v4f __attribute__((ext_vector_type(4)));
typedef float v8f __attribute__((ext_vector_type(8)));

#if defined(__has_builtin)
#  if __has_builtin(__builtin_amdgcn_global_load_async_to_lds_b32)
#    define HAVE_ASYNC_LDS 1
#  endif
#  if __has_builtin(__builtin_amdgcn_s_wait_asynccnt)
#    define HAVE_WAIT_ASYNC 1
#  endif
#endif
#ifndef HAVE_ASYNC_LDS
#  define HAVE_ASYNC_LDS 0
#endif
#ifndef HAVE_WAIT_ASYNC
#  define HAVE_WAIT_ASYNC 0
#endif

__device__ __forceinline__ void wait_asynccnt0() {
#if HAVE_WAIT_ASYNC
    __builtin_amdgcn_s_wait_asynccnt(0);
#else
    asm volatile("s_wait_asynccnt 0x0" ::: "memory");
#endif
}

// ---------------- three_nn: top-3 nearest sources per target ----------------
__global__ __launch_bounds__(256) void nn_kernel(
    const float* __restrict__ target, const float* __restrict__ source,
    float* __restrict__ wgt, int* __restrict__ idx)
{
    __shared__ float sx[MPTS], sy[MPTS], sz[MPTS];
    const int b = blockIdx.y;
    const int tid = threadIdx.x;
    for (int i = tid; i < MPTS; i += 256) {
        const float* sp = source + ((size_t)b * MPTS + i) * 3;
        sx[i] = sp[0]; sy[i] = sp[1]; sz[i] = sp[2];
    }
    __syncthreads();
    const int p = blockIdx.x * 256 + tid;
    const float* tp = target + ((size_t)b * NPTS + p) * 3;
    const float tx = tp[0], ty = tp[1], tz = tp[2];
    float d0 = 3.4e38f, d1 = 3.4e38f, d2 = 3.4e38f;
    int   i0 = 0, i1 = 0, i2 = 0;
    for (int j = 0; j < MPTS; ++j) {
        const float dx = tx - sx[j], dy = ty - sy[j], dz = tz - sz[j];
        const float d = dx * dx + dy * dy + dz * dz;
        if (d < d0)      { d2 = d1; i2 = i1; d1 = d0; i1 = i0; d0 = d; i0 = j; }
        else if (d < d1) { d2 = d1; i2 = i1; d1 = d;  i1 = j; }
        else if (d < d2) { d2 = d;  i2 = j; }
    }
    const float r0 = 1.f / (d0 + 1e-8f);
    const float r1 = 1.f / (d1 + 1e-8f);
    const float r2 = 1.f / (d2 + 1e-8f);
    const float inv = 1.f / (r0 + r1 + r2);
    const size_t o = ((size_t)b * NPTS + p) * 3;
    wgt[o] = r0 * inv; wgt[o + 1] = r1 * inv; wgt[o + 2] = r2 * inv;
    idx[o] = i0;       idx[o + 1] = i1;       idx[o + 2] = i2;
}

// -------- transpose source_feats (B,C2,m) -> (B,m,C2), 32x32 LDS tiles ------
__global__ __launch_bounds__(256) void transpose_sf(
    const float* __restrict__ sf, float* __restrict__ sfT)
{
    __shared__ float tile[32][33];
    const int b = blockIdx.z;
    const int m0 = blockIdx.x * 32, c0 = blockIdx.y * 32;
    const int tx = threadIdx.x & 31, ty = threadIdx.x >> 5;  // 32 x 8
#pragma unroll
    for (int k = 0; k < 4; ++k)
        tile[ty + 8 * k][tx] = sf[((size_t)b * C2 + c0 + ty + 8 * k) * MPTS + m0 + tx];
    __syncthreads();
#pragma unroll
    for (int k = 0; k < 4; ++k)
        sfT[((size_t)b * MPTS + m0 + ty + 8 * k) * C2 + c0 + tx] = tile[tx][ty + 8 * k];
}

// -------------------- BN eval-mode scale/bias precompute --------------------
__global__ void scalebias_kernel(
    const float* __restrict__ g0, const float* __restrict__ b0,
    const float* __restrict__ mu0, const float* __restrict__ var0,
    const float* __restrict__ g1, const float* __restrict__ b1,
    const float* __restrict__ mu1, const float* __restrict__ var1,
    float* __restrict__ sb)
{
    const int t = threadIdx.x;  // 256 threads
    const float s0 = g0[t] * rsqrtf(var0[t] + 1e-5f);
    sb[t]       = s0;
    sb[256 + t] = b0[t] - mu0[t] * s0;
    const float s1 = g1[t] * rsqrtf(var1[t] + 1e-5f);
    sb[512 + t] = s1;
    sb[768 + t] = b1[t] - mu1[t] * s1;
}

// BN + ReLU on a 16x16 f32 accumulator fragment, then 2x ds_store_b128.
__device__ __forceinline__ void bn_relu_store_lds(
    const v8f& acc, const float* __restrict__ sc, const float* __restrict__ bi,
    float* __restrict__ dst)
{
    v4f lo, hi;
#pragma unroll
    for (int r = 0; r < 4; ++r) {
        const float v = acc[r] * sc[r] + bi[r];
        lo[r] = v > 0.f ? v : 0.f;
    }
#pragma unroll
    for (int r = 4; r < 8; ++r) {
        const float v = acc[r] * sc[r] + bi[r];
        hi[r - 4] = v > 0.f ? v : 0.f;
    }
    *(v4f*)dst       = lo;
    *(v4f*)(dst + 4) = hi;
}

// ---- fused: interp+concat (LDS) -> GEMM0+BN+ReLU (LDS) -> GEMM1+BN+ReLU ----
__global__ __launch_bounds__(256) void fp_fused_kernel(
    const float* __restrict__ tf,    // target_feats (B,C1,n)
    const float* __restrict__ w0,    // (256,384) row-major
    const float* __restrict__ w1,    // (256,256) row-major
    const float* __restrict__ sfT,   // (B,m,C2)
    const float* __restrict__ wgt, const int* __restrict__ idx,
    const float* __restrict__ sb,    // scale0|bias0|scale1|bias1 (256 each)
    float* __restrict__ out)         // (B,256,n)
{
    extern __shared__ float smem[];
    float* Xt  = smem;                      // NCOLS * XSTR  (later reused as Y1t)
    float* Y0t = Xt + NCOLS * XSTR;         // NCOLS * YSTR
    float* wq  = Y0t + NCOLS * YSTR;        // NCOLS * 3
    int*   iq  = (int*)(wq + NCOLS * 3);    // NCOLS * 3

    const int b   = blockIdx.y;
    const int n0  = blockIdx.x * NCOLS;
    const int tid = threadIdx.x;

    // stage per-column 3-NN idx/weights
    if (tid < NCOLS * 3) {
        const size_t o = ((size_t)b * NPTS + n0) * 3 + tid;
        wq[tid] = wgt[o];
        iq[tid] = idx[o];
    }

    // Xt rows 256..383 <- target_feats (async global -> LDS, ASYNCcnt-tracked)
    {
        const float* tfb = tf + (size_t)b * C1 * NPTS + n0;
        for (int e = tid; e < C1 * NCOLS; e += 256) {
            const int ch = e >> 5;          // NCOLS == 32
            const int c  = e & (NCOLS - 1);
#if HAVE_ASYNC_LDS
            __builtin_amdgcn_global_load_async_to_lds_b32(
                (int*)(tfb + (size_t)ch * NPTS + c),
                (int*)(Xt + c * XSTR + C2 + ch),
                /*offset=*/0, /*cpol=*/0);
#else
            Xt[c * XSTR + C2 + ch] = tfb[(size_t)ch * NPTS + c];
#endif
        }
    }
    __syncthreads();   // wq/iq visible

    // Xt rows 0..255 <- interpolated source features (coalesced gathers)
    {
        const float* sfb = sfT + (size_t)b * MPTS * C2;
        const int ch = tid;  // one channel per thread
        for (int c = 0; c < NCOLS; ++c) {
            const float v = wq[c * 3 + 0] * sfb[(size_t)iq[c * 3 + 0] * C2 + ch]
                          + wq[c * 3 + 1] * sfb[(size_t)iq[c * 3 + 1] * C2 + ch]
                          + wq[c * 3 + 2] * sfb[(size_t)iq[c * 3 + 2] * C2 + ch];
            Xt[c * XSTR + ch] = v;
        }
    }
#if HAVE_ASYNC_LDS
    wait_asynccnt0();  // async target_feats tiles landed in LDS
#endif
    __syncthreads();

    const int lane  = tid & 31;
    const int wave  = tid >> 5;         // 0..7
    const int l15   = lane & 15;
    const int khalf = lane >> 4;        // A/B fragment K-half select
    const int nt    = wave & 1;         // N-tile within slab
    const int cn    = nt * 16;
    const int mtb   = (wave >> 1) * 4;  // 4 consecutive M-tiles per wave
    const int rsel  = 8 * khalf;        // row sub-block within fragment

    const float* xcol = Xt + (cn + l15) * XSTR + 2 * khalf;

    // ---------------- GEMM0: Y0 = relu(bn(W0 @ X)) ----------------
    {
        const float* a0p = w0 + (size_t)((mtb + 0) * 16 + l15) * CIN + 2 * khalf;
        const float* a1p = w0 + (size_t)((mtb + 1) * 16 + l15) * CIN + 2 * khalf;
        const float* a2p = w0 + (size_t)((mtb + 2) * 16 + l15) * CIN + 2 * khalf;
        const float* a3p = w0 + (size_t)((mtb + 3) * 16 + l15) * CIN + 2 * khalf;
        v8f c0 = {0.f,0.f,0.f,0.f,0.f,0.f,0.f,0.f};
        v8f c1 = c0, c2 = c0, c3 = c0;
#pragma unroll 4
        for (int k = 0; k < CIN; k += 4) {
            const v2f bf = *(const v2f*)(xcol + k);   // one B fragment ...
            c0 = __builtin_amdgcn_wmma_f32_16x16x4_f32(false, *(const v2f*)(a0p + k),
                     false, bf, (short)0, c0, false, false);   // ... feeds 4 WMMAs
            c1 = __builtin_amdgcn_wmma_f32_16x16x4_f32(false, *(const v2f*)(a1p + k),
                     false, bf, (short)0, c1, false, false);
            c2 = __builtin_amdgcn_wmma_f32_16x16x4_f32(false, *(const v2f*)(a2p + k),
                     false, bf, (short)0, c2, false, false);
            c3 = __builtin_amdgcn_wmma_f32_16x16x4_f32(false, *(const v2f*)(a3p + k),
                     false, bf, (short)0, c3, false, false);
        }
        float* ycol = Y0t + (cn + l15) * YSTR;
        const int r0 = (mtb + 0) * 16 + rsel;
        const int r1 = (mtb + 1) * 16 + rsel;
        const int r2 = (mtb + 2) * 16 + rsel;
        const int r3 = (mtb + 3) * 16 + rsel;
        bn_relu_store_lds(c0, sb + r0, sb + 256 + r0, ycol + r0);
        bn_relu_store_lds(c1, sb + r1, sb + 256 + r1, ycol + r1);
        bn_relu_store_lds(c2, sb + r2, sb + 256 + r2, ycol + r2);
        bn_relu_store_lds(c3, sb + r3, sb + 256 + r3, ycol + r3);
    }
    __syncthreads();

    // ---------------- GEMM1: Y1 = relu(bn(W1 @ Y0)) (staged in LDS) --------
    {
        const float* ycol = Y0t + (cn + l15) * YSTR + 2 * khalf;
        const float* a0p = w1 + (size_t)((mtb + 0) * 16 + l15) * CO + 2 * khalf;
        const float* a1p = w1 + (size_t)((mtb + 1) * 16 + l15) * CO + 2 * khalf;
        const float* a2p = w1 + (size_t)((mtb + 2) * 16 + l15) * CO + 2 * khalf;
        const float* a3p = w1 + (size_t)((mtb + 3) * 16 + l15) * CO + 2 * khalf;
        v8f c0 = {0.f,0.f,0.f,0.f,0.f,0.f,0.f,0.f};
        v8f c1 = c0, c2 = c0, c3 = c0;
#pragma unroll 4
        for (int k = 0; k < CO; k += 4) {
            const v2f bf = *(const v2f*)(ycol + k);
            c0 = __builtin_amdgcn_wmma_f32_16x16x4_f32(false, *(const v2f*)(a0p + k),
                     false, bf, (short)0, c0, false, false);
            c1 = __builtin_amdgcn_wmma_f32_16x16x4_f32(false, *(const v2f*)(a1p + k),
                     false, bf, (short)0, c1, false, false);
            c2 = __builtin_amdgcn_wmma_f32_16x16x4_f32(false, *(const v2f*)(a2p + k),
                     false, bf, (short)0, c2, false, false);
            c3 = __builtin_amdgcn_wmma_f32_16x16x4_f32(false, *(const v2f*)(a3p + k),
                     false, bf, (short)0, c3, false, false);
        }
        float* y1col = Xt + (cn + l15) * YSTR;   // reuse Xt region as Y1t
        const int r0 = (mtb + 0) * 16 + rsel;
        const int r1 = (mtb + 1) * 16 + rsel;
        const int r2 = (mtb + 2) * 16 + rsel;
        const int r3 = (mtb + 3) * 16 + rsel;
        bn_relu_store_lds(c0, sb + 512 + r0, sb + 768 + r0, y1col + r0);
        bn_relu_store_lds(c1, sb + 512 + r1, sb + 768 + r1, y1col + r1);
        bn_relu_store_lds(c2, sb + 512 + r2, sb + 768 + r2, y1col + r2);
        bn_relu_store_lds(c3, sb + 512 + r3, sb + 768 + r3, y1col + r3);
    }
    __syncthreads();

    // -------- coalesced output: one 128B row-segment per wave per iter ------
    {
        const float* Y1t = Xt;
        float* ob = out + (size_t)b * CO * NPTS + n0;
        for (int e = tid; e < CO * NCOLS; e += 256) {
            const int row = e >> 5;          // NCOLS == 32
            const int c   = e & (NCOLS - 1);
            ob[(size_t)row * NPTS + c] = Y1t[c * YSTR + row];
        }
    }
}

// ---------------------------------------------------------------------------
extern "C" void kernel_launch(void* const* d_in, const int* in_sizes, int n_in,
                              void* d_out, int out_size, void* d_ws, size_t ws_size,
                              hipStream_t stream) {
    const float* target = (const float*)d_in[0];
    const float* source = (const float*)d_in[1];
    const float* tfeats = (const float*)d_in[2];
    const float* sfeats = (const float*)d_in[3];
    const float* w0   = (const float*)d_in[4];
    const float* g0   = (const float*)d_in[5];
    const float* b0   = (const float*)d_in[6];
    const float* mu0  = (const float*)d_in[7];
    const float* var0 = (const float*)d_in[8];
    const float* w1   = (const float*)d_in[9];
    const float* g1   = (const float*)d_in[10];
    const float* b1   = (const float*)d_in[11];
    const float* mu1  = (const float*)d_in[12];
    const float* var1 = (const float*)d_in[13];
    float* out = (float*)d_out;

    // workspace layout (floats): sfT | wgt | idx | scale/bias   (~11.5 MB)
    float* sfT = (float*)d_ws;
    float* wgt = sfT + (size_t)B_ * MPTS * C2;
    int*   idx = (int*)(wgt + (size_t)B_ * NPTS * 3);
    float* sb  = (float*)(idx + (size_t)B_ * NPTS * 3);

    transpose_sf<<<dim3(MPTS / 32, C2 / 32, B_), 256, 0, stream>>>(sfeats, sfT);
    scalebias_kernel<<<1, 256, 0, stream>>>(g0, b0, mu0, var0, g1, b1, mu1, var1, sb);
    nn_kernel<<<dim3(NPTS / 256, B_), 256, 0, stream>>>(target, source, wgt, idx);

    const size_t lds_bytes =
        (size_t)(NCOLS * XSTR + NCOLS * YSTR + NCOLS * 3) * sizeof(float) +
        (size_t)NCOLS * 3 * sizeof(int);   // 83,712 B
    fp_fused_kernel<<<dim3(NPTS / NCOLS, B_), 256, lds_bytes, stream>>>(
        tfeats, w0, w1, sfT, wgt, idx, sb, out);
}